// FourBranchFork_64407329571545
// MI455X (gfx1250) — compile-verified
//
#include <hip/hip_runtime.h>

typedef float v2f __attribute__((ext_vector_type(2)));
typedef float v4f __attribute__((ext_vector_type(4)));
typedef float v8f __attribute__((ext_vector_type(8)));
typedef int   v4i __attribute__((vector_size(16)));

#define __AS1 __attribute__((address_space(1)))
#define __AS3 __attribute__((address_space(3)))

#define TILE_M 64
#define TILE_N 64
#define TILE_K 32
#define APAD 4           // row pad for [row][k]-style LDS tiles (stride 36 floats)
#define LN_EPS 1e-5f

// ---------------------------------------------------------------------------
// Async global->LDS copy (CDNA5 GLOBAL_LOAD_ASYNC_TO_LDS_B128, ASYNCcnt),
// with a synchronous global_load_b128 + ds_store fallback (other toolchain).
// ---------------------------------------------------------------------------
#if __has_builtin(__builtin_amdgcn_global_load_async_to_lds_b128)
#define USE_ASYNC 1
#else
#define USE_ASYNC 0
#endif

#if USE_ASYNC
#if __has_builtin(__builtin_amdgcn_s_wait_asynccnt)
#define ASYNC_WAIT(n) __builtin_amdgcn_s_wait_asynccnt(n)
#else
#define ASYNC_WAIT(n) asm volatile("s_wait_asynccnt %0" ::"i"(n) : "memory")
#endif
#else
#define ASYNC_WAIT(n) ((void)0)
#endif

__device__ __forceinline__ void cp16(float* lds, const float* g) {
#if USE_ASYNC
  __builtin_amdgcn_global_load_async_to_lds_b128(
      (__AS1 v4i*)g, (__AS3 v4i*)lds, /*offset=*/0, /*cpol=*/0);
#else
  *(v4f*)lds = *(const __AS1 v4f*)g;
#endif
}

__device__ __forceinline__ v8f wmma_f32(v2f a, v2f b, v8f c) {
  // D = A(16x4 f32) x B(4x16 f32) + C(16x16 f32)
  return __builtin_amdgcn_wmma_f32_16x16x4_f32(
      /*neg_a=*/false, a, /*neg_b=*/false, b,
      /*c_mod=*/(short)0, c, /*reuse_a=*/false, /*reuse_b=*/false);
}

// ---------------------------------------------------------------------------
// Phase A: E[u][v] = sum_br sum_o D[u][o] * W[o][v] * S[o/128][v/128]
// D: [H][K] row-major, W: [K][H] row-major, S: [K/128][H/128]
// Branchless double-buffered async pipeline; block scale folded into the A
// fragment (uniform scalar per (branch, k0): 64-wide N tile and 32-deep K
// tile are both 128-block aligned).
// ---------------------------------------------------------------------------
__global__ __launch_bounds__(256)
void build_fused_weight(const float* __restrict__ d0, const float* __restrict__ w0,
                        const float* __restrict__ s0, int K0,
                        const float* __restrict__ d1, const float* __restrict__ w1,
                        const float* __restrict__ s1, int K1,
                        const float* __restrict__ d2, const float* __restrict__ w2,
                        const float* __restrict__ s2, int K2,
                        const float* __restrict__ d3, const float* __restrict__ w3,
                        const float* __restrict__ s3, int K3,
                        float* __restrict__ E, int H) {
  __shared__ __attribute__((aligned(16))) float As[2][TILE_M][TILE_K + APAD];
  __shared__ __attribute__((aligned(16))) float Bs[2][TILE_K][TILE_N];

  const int tid  = threadIdx.x;
  const int lane = tid & 31;
  const int wave = tid >> 5;     // 8 waves: 4 (M) x 2 (N)
  const int wm   = wave >> 1;    // 0..3
  const int wn   = wave & 1;     // 0..1
  const int hi   = (lane >> 4);  // upper half-wave
  const int l    = lane & 15;

  const int m0g = blockIdx.y * TILE_M;
  const int n0g = blockIdx.x * TILE_N;

  const float* Dp[4] = {d0, d1, d2, d3};
  const float* Wp[4] = {w0, w1, w2, w3};
  const float* Sp[4] = {s0, s1, s2, s3};
  const int    Kp[4] = {K0, K1, K2, K3};

  v8f acc0 = {};
  v8f acc1 = {};

  const int lm  = tid & 63;   // A staging: row m (0..63)
  const int lk4 = tid >> 6;   // A staging: 16B chunk (0..3), +4 for second half
  const int bk  = tid >> 4;   // B staging: k row (0..15), +16 for second half
  const int bn4 = tid & 15;   // B staging: 16B chunk (0..15)

  const int s_ld = H >> 7;    // scale row stride = H/128
  const int snb  = n0g >> 7;  // scale column block

  const int am  = wm * 16 + l;
  const int nb0 = wn * 32 + l;

  for (int br = 0; br < 4; ++br) {
    const float* __restrict__ A = Dp[br];
    const float* __restrict__ B = Wp[br];
    const float* __restrict__ S = Sp[br];
    const int K = Kp[br];

    const float* Arow = &A[(size_t)(m0g + lm) * K];

    // Pipeline prologue: stage tile k0=0 into buffer 0 (4 B128 copies/thread).
    cp16(&As[0][lm][lk4 * 4],       &Arow[lk4 * 4]);
    cp16(&As[0][lm][lk4 * 4 + 16],  &Arow[lk4 * 4 + 16]);
    cp16(&Bs[0][bk][bn4 * 4],       &B[(size_t)bk * H + n0g + bn4 * 4]);
    cp16(&Bs[0][bk + 16][bn4 * 4],  &B[(size_t)(bk + 16) * H + n0g + bn4 * 4]);

    int buf = 0;
    for (int k0 = 0; k0 < K; k0 += TILE_K) {
      // Branchless: last iteration redundantly re-stages the current tile
      // into the spare buffer (never consumed; drained below).
      const int kn = (k0 + TILE_K < K) ? (k0 + TILE_K) : k0;
      cp16(&As[buf ^ 1][lm][lk4 * 4],      &Arow[kn + lk4 * 4]);
      cp16(&As[buf ^ 1][lm][lk4 * 4 + 16], &Arow[kn + lk4 * 4 + 16]);
      cp16(&Bs[buf ^ 1][bk][bn4 * 4],
           &B[(size_t)(kn + bk) * H + n0g + bn4 * 4]);
      cp16(&Bs[buf ^ 1][bk + 16][bn4 * 4],
           &B[(size_t)(kn + bk + 16) * H + n0g + bn4 * 4]);

      // 8 in flight (current 4 + next 4); <=4 means current tile landed.
      ASYNC_WAIT(4);
      __syncthreads();

      const float sv = *(const __AS1 float*)&S[(k0 >> 7) * s_ld + snb];

#pragma unroll
      for (int kk = 0; kk < TILE_K; kk += 4) {
        v2f a;
        a.x = As[buf][am][kk + (hi ? 2 : 0)] * sv;
        a.y = As[buf][am][kk + (hi ? 3 : 1)] * sv;
        v2f b0, b1;
        b0.x = Bs[buf][kk + hi][nb0];
        b0.y = Bs[buf][kk + 2 + hi][nb0];
        b1.x = Bs[buf][kk + hi][nb0 + 16];
        b1.y = Bs[buf][kk + 2 + hi][nb0 + 16];
        acc0 = wmma_f32(a, b0, acc0);
        acc1 = wmma_f32(a, b1, acc1);
      }
      __syncthreads();
      buf ^= 1;
    }

    // Drain leftover redundant copies before the next branch reuses buffers.
    ASYNC_WAIT(0);
    __syncthreads();
  }

  // Epilogue: C vgpr r -> rows (r, r+8); lanes 0-15 / 16-31 split M by 8.
  const int row0 = m0g + wm * 16 + hi * 8;
  const int col0 = n0g + wn * 32 + l;
#pragma unroll
  for (int r = 0; r < 8; ++r) {
    *(__AS1 float*)&E[(size_t)(row0 + r) * H + col0]      = acc0[r];
    *(__AS1 float*)&E[(size_t)(row0 + r) * H + col0 + 16] = acc1[r];
  }
}

// ---------------------------------------------------------------------------
// Phase B: h[n][u] = sum_k x[n][k] * E[u][k]   (C = A * B^T, both K-fast)
// ---------------------------------------------------------------------------
__global__ __launch_bounds__(256)
void gemm_x_Et(const float* __restrict__ x, const float* __restrict__ E,
               float* __restrict__ h, int H) {
  __shared__ __attribute__((aligned(16))) float As[2][TILE_M][TILE_K + APAD];
  __shared__ __attribute__((aligned(16))) float Es[2][TILE_N][TILE_K + APAD];

  const int tid  = threadIdx.x;
  const int lane = tid & 31;
  const int wave = tid >> 5;
  const int wm   = wave >> 1;
  const int wn   = wave & 1;
  const int hi   = (lane >> 4);
  const int l    = lane & 15;

  const int m0g = blockIdx.y * TILE_M;   // token rows
  const int n0g = blockIdx.x * TILE_N;   // output feature cols

  const int lm  = tid & 63;   // x staging: row (0..63)
  const int lk4 = tid >> 6;   // x staging: 16B chunk (0..3), +4 for 2nd half
  const int en  = tid >> 2;   // E staging: row (0..63)
  const int ek4 = tid & 3;    // E staging: 16B chunk (0..3), +4 for 2nd half

  const int am  = wm * 16 + l;
  const int nb0 = wn * 32 + l;

  v8f acc0 = {};
  v8f acc1 = {};

  const float* xrow = &x[(size_t)(m0g + lm) * H];
  const float* erow = &E[(size_t)(n0g + en) * H];

  // Pipeline prologue
  cp16(&As[0][lm][lk4 * 4],      &xrow[lk4 * 4]);
  cp16(&As[0][lm][lk4 * 4 + 16], &xrow[lk4 * 4 + 16]);
  cp16(&Es[0][en][ek4 * 4],      &erow[ek4 * 4]);
  cp16(&Es[0][en][ek4 * 4 + 16], &erow[ek4 * 4 + 16]);

  int buf = 0;
  for (int k0 = 0; k0 < H; k0 += TILE_K) {
    const int kn = (k0 + TILE_K < H) ? (k0 + TILE_K) : k0;
    cp16(&As[buf ^ 1][lm][lk4 * 4],      &xrow[kn + lk4 * 4]);
    cp16(&As[buf ^ 1][lm][lk4 * 4 + 16], &xrow[kn + lk4 * 4 + 16]);
    cp16(&Es[buf ^ 1][en][ek4 * 4],      &erow[kn + ek4 * 4]);
    cp16(&Es[buf ^ 1][en][ek4 * 4 + 16], &erow[kn + ek4 * 4 + 16]);

    ASYNC_WAIT(4);
    __syncthreads();

#pragma unroll
    for (int kk = 0; kk < TILE_K; kk += 4) {
      v2f a;
      a.x = As[buf][am][kk + (hi ? 2 : 0)];
      a.y = As[buf][am][kk + (hi ? 3 : 1)];
      v2f b0, b1;
      b0.x = Es[buf][nb0][kk + hi];
      b0.y = Es[buf][nb0][kk + 2 + hi];
      b1.x = Es[buf][nb0 + 16][kk + hi];
      b1.y = Es[buf][nb0 + 16][kk + 2 + hi];
      acc0 = wmma_f32(a, b0, acc0);
      acc1 = wmma_f32(a, b1, acc1);
    }
    __syncthreads();
    buf ^= 1;
  }

  const int row0 = m0g + wm * 16 + hi * 8;
  const int col0 = n0g + wn * 32 + l;
#pragma unroll
  for (int r = 0; r < 8; ++r) {
    *(__AS1 float*)&h[(size_t)(row0 + r) * H + col0]      = acc0[r];
    *(__AS1 float*)&h[(size_t)(row0 + r) * H + col0 + 16] = acc1[r];
  }
}

// ---------------------------------------------------------------------------
// Phase C: row LayerNorm in-place on h (one block per row, 128-bit accesses)
// ---------------------------------------------------------------------------
__global__ __launch_bounds__(256)
void layernorm_kernel(float* __restrict__ h, const float* __restrict__ gamma,
                      const float* __restrict__ beta, int H) {
  __shared__ float r1[256];
  __shared__ float r2[256];

  const int row = blockIdx.x;
  float* __restrict__ p = h + (size_t)row * H;
  const int HV = H >> 2;  // 128-bit elements per row

  float s = 0.f, sq = 0.f;
  for (int c = threadIdx.x; c < HV; c += 256) {
    v4f v = *(const __AS1 v4f*)&p[c * 4];
    s  += v.x + v.y + v.z + v.w;
    sq += v.x * v.x + v.y * v.y + v.z * v.z + v.w * v.w;
  }
  r1[threadIdx.x] = s;
  r2[threadIdx.x] = sq;
  __syncthreads();
  for (int off = 128; off > 0; off >>= 1) {
    if (threadIdx.x < off) {
      r1[threadIdx.x] += r1[threadIdx.x + off];
      r2[threadIdx.x] += r2[threadIdx.x + off];
    }
    __syncthreads();
  }
  const float mean = r1[0] / (float)H;
  const float var  = r2[0] / (float)H - mean * mean;
  const float rstd = rsqrtf(var + LN_EPS);

  for (int c = threadIdx.x; c < HV; c += 256) {
    v4f v = *(const __AS1 v4f*)&p[c * 4];
    v4f g = *(const __AS1 v4f*)&gamma[c * 4];
    v4f b = *(const __AS1 v4f*)&beta[c * 4];
    v4f o;
    o.x = (v.x - mean) * rstd * g.x + b.x;
    o.y = (v.y - mean) * rstd * g.y + b.y;
    o.z = (v.z - mean) * rstd * g.z + b.z;
    o.w = (v.w - mean) * rstd * g.w + b.w;
    *(__AS1 v4f*)&p[c * 4] = o;
  }
}

// ---------------------------------------------------------------------------
extern "C" void kernel_launch(void* const* d_in, const int* in_sizes, int n_in,
                              void* d_out, int out_size, void* d_ws, size_t ws_size,
                              hipStream_t stream) {
  const float* x     = (const float*)d_in[0];
  const float* w_qkv = (const float*)d_in[1];
  const float* s_qkv = (const float*)d_in[2];
  const float* w_z   = (const float*)d_in[3];
  const float* s_z   = (const float*)d_in[4];
  const float* w_b   = (const float*)d_in[5];
  const float* s_b   = (const float*)d_in[6];
  const float* w_a   = (const float*)d_in[7];
  const float* s_a   = (const float*)d_in[8];
  const float* d_qkv = (const float*)d_in[9];
  const float* d_z   = (const float*)d_in[10];
  const float* d_b   = (const float*)d_in[11];
  const float* d_a   = (const float*)d_in[12];
  const float* gamma = (const float*)d_in[13];
  const float* beta  = (const float*)d_in[14];

  const int H    = in_sizes[13];           // 2048
  const int Ntok = in_sizes[0] / H;        // 8192
  const int L    = in_sizes[1] / H;        // 8192
  const int M    = in_sizes[3] / H;        // 4096
  const int S    = in_sizes[5] / H;        // 1024

  float* E = (float*)d_ws;                 // H*H fp32 fused weight (16 MB)
  float* h = (float*)d_out;                // [Ntok, H] output / LN buffer

  dim3 blk(256);

  dim3 gridE(H / TILE_N, H / TILE_M);
  build_fused_weight<<<gridE, blk, 0, stream>>>(
      d_qkv, w_qkv, s_qkv, L,
      d_z,   w_z,   s_z,   M,
      d_b,   w_b,   s_b,   S,
      d_a,   w_a,   s_a,   S,
      E, H);

  dim3 gridH(H / TILE_N, Ntok / TILE_M);
  gemm_x_Et<<<gridH, blk, 0, stream>>>(x, E, h, H);

  layernorm_kernel<<<Ntok, blk, 0, stream>>>(h, gamma, beta, H);
}